// SocialGCN_12025908429029
// MI455X (gfx1250) — compile-verified
//
#include <hip/hip_runtime.h>

#define WPB 8      // waves per block (256 threads, wave32)
#define BLOCKT 256

typedef unsigned int uint32;

// ---- CDNA5 async-to-LDS path (ASYNCcnt), guarded so compile never breaks ----
#if __has_builtin(__builtin_amdgcn_global_load_async_to_lds_b32) && \
    __has_builtin(__builtin_amdgcn_s_wait_asynccnt)
#define USE_ASYNC 1
#else
#define USE_ASYNC 0
#endif

#if USE_ASYNC
// builtin signature: (AS1 int* src, AS3 int* dst, imm offset, imm cpol)
typedef __attribute__((address_space(1))) int GInt;
typedef __attribute__((address_space(3))) int LInt;
__device__ __forceinline__ GInt* to_global(const void* p) {
  return (GInt*)(unsigned long long)p;           // int round-trip: AS reinterpret
}
__device__ __forceinline__ LInt* to_lds(const void* p) {
  // generic LDS pointer: low 32 bits are the LDS byte offset (flat LDS aperture)
  return (LInt*)(unsigned int)(unsigned long long)p;
}
#endif

__device__ __forceinline__ void async_wait_all() {
#if USE_ASYNC
  __builtin_amdgcn_s_wait_asynccnt(0);
#endif
  __asm__ volatile("" ::: "memory");
}

// ---------------- main hop kernel (first in file so disasm snippet shows it) --
// One wave per row. 32 lanes x float4 = one 128-float row.
// Edge (col,val) chunks staged into LDS with double-buffered async-to-LDS.
// Row index / extents scalarized via readfirstlane (wave-uniform by construction)
// so row_ptr loads are SMEM, chunk-loop control is SALU, gathers are SADDR-form.
// Epilogue: dst[r] = row result (if dst != null), acc[r] += 0.25f * row result.
__global__ __launch_bounds__(BLOCKT) void spmm_csr_kernel(
    const uint32* __restrict__ row_ptr, const uint32* __restrict__ csr_col,
    const float* __restrict__ csr_val, const float* __restrict__ src,
    float* __restrict__ dst, float* __restrict__ acc, int n_rows) {
  __shared__ uint32 s_col[2][WPB][32];
  __shared__ float  s_val[2][WPB][32];
  const int lane = threadIdx.x & 31;
  const int ws   = threadIdx.x >> 5;
  const int wave = blockIdx.x * WPB + ws;
  const int nw   = gridDim.x * WPB;
  const float4* __restrict__ src4 = (const float4*)src;

  for (int rr = wave; rr < n_rows; rr += nw) {
    const int r = __builtin_amdgcn_readfirstlane(rr);            // scalar row id
    const uint32 start = row_ptr[r];                             // uniform -> s_load
    const uint32 end   = row_ptr[r + 1];

    auto issue = [&](int buf, uint32 base) {
      uint32 e = base + (uint32)lane;
      if (e < end) {
#if USE_ASYNC
        __builtin_amdgcn_global_load_async_to_lds_b32(to_global(csr_col + e),
                                                      to_lds(&s_col[buf][ws][lane]), 0, 0);
        __builtin_amdgcn_global_load_async_to_lds_b32(to_global(csr_val + e),
                                                      to_lds(&s_val[buf][ws][lane]), 0, 0);
#else
        s_col[buf][ws][lane] = csr_col[e];
        s_val[buf][ws][lane] = csr_val[e];
#endif
      }
    };

    // two accumulators: halves the FMA dependency chain, doubles gather MLP
    float4 a0 = make_float4(0.f, 0.f, 0.f, 0.f);
    float4 a1 = make_float4(0.f, 0.f, 0.f, 0.f);
    int cur = 0;
    issue(0, start);
    async_wait_all();

    for (uint32 e0 = start; e0 < end; e0 += 32u) {
      uint32 nxt = e0 + 32u;
      if (nxt < end) issue(cur ^ 1, nxt);           // prefetch next chunk
      const int m = (int)(((end - e0) < 32u) ? (end - e0) : 32u);
      int i = 0;
      for (; i + 1 < m; i += 2) {
        int   c0 = __builtin_amdgcn_readfirstlane((int)s_col[cur][ws][i]);
        float v0 = __int_as_float(__builtin_amdgcn_readfirstlane(__float_as_int(s_val[cur][ws][i])));
        int   c1 = __builtin_amdgcn_readfirstlane((int)s_col[cur][ws][i + 1]);
        float v1 = __int_as_float(__builtin_amdgcn_readfirstlane(__float_as_int(s_val[cur][ws][i + 1])));
        float4 x0 = src4[(size_t)(uint32)c0 * 32u + (uint32)lane];  // L2-resident gather
        float4 x1 = src4[(size_t)(uint32)c1 * 32u + (uint32)lane];
        a0.x = fmaf(v0, x0.x, a0.x); a0.y = fmaf(v0, x0.y, a0.y);
        a0.z = fmaf(v0, x0.z, a0.z); a0.w = fmaf(v0, x0.w, a0.w);
        a1.x = fmaf(v1, x1.x, a1.x); a1.y = fmaf(v1, x1.y, a1.y);
        a1.z = fmaf(v1, x1.z, a1.z); a1.w = fmaf(v1, x1.w, a1.w);
      }
      if (i < m) {
        int   c = __builtin_amdgcn_readfirstlane((int)s_col[cur][ws][i]);
        float v = __int_as_float(__builtin_amdgcn_readfirstlane(__float_as_int(s_val[cur][ws][i])));
        float4 x = src4[(size_t)(uint32)c * 32u + (uint32)lane];
        a0.x = fmaf(v, x.x, a0.x); a0.y = fmaf(v, x.y, a0.y);
        a0.z = fmaf(v, x.z, a0.z); a0.w = fmaf(v, x.w, a0.w);
      }
      async_wait_all();
      cur ^= 1;
    }

    float4 a = make_float4(a0.x + a1.x, a0.y + a1.y, a0.z + a1.z, a0.w + a1.w);
    const size_t o = (size_t)r * 32u + (uint32)lane;
    if (dst) ((float4*)dst)[o] = a;                 // plain store, no atomics
    float4 t = ((float4*)acc)[o];                   // wave exclusively owns row r
    t.x += 0.25f * a.x; t.y += 0.25f * a.y; t.z += 0.25f * a.z; t.w += 0.25f * a.w;
    ((float4*)acc)[o] = t;
  }
}

// ---------------- small utility kernels ----------------

__global__ void zero_u32_kernel(uint32* __restrict__ p, int n) {
  int i = blockIdx.x * blockDim.x + threadIdx.x;
  int stride = gridDim.x * blockDim.x;
  for (; i < n; i += stride) p[i] = 0u;
}

// out = 0.25f * users_emb  (folds the /(HOP+1) for hop 0)
__global__ void init_out_kernel(const float4* __restrict__ in, float4* __restrict__ out, int n4) {
  int i = blockIdx.x * blockDim.x + threadIdx.x;
  int stride = gridDim.x * blockDim.x;
  for (; i < n4; i += stride) {
    float4 a = in[i];
    out[i] = make_float4(0.25f * a.x, 0.25f * a.y, 0.25f * a.z, 0.25f * a.w);
  }
}

// per-row degree histogram (int64 indices -> low dword, N < 2^31)
__global__ void hist_kernel(const long long* __restrict__ erow, uint32* __restrict__ cnt, int ne) {
  int i = blockIdx.x * blockDim.x + threadIdx.x;
  int stride = gridDim.x * blockDim.x;
  for (; i < ne; i += stride) atomicAdd(&cnt[(uint32)erow[i]], 1u);
}

// single-block exclusive scan (N=100k -> ~98 chunks of 1024); also zeroes cnt
// so it can be reused as the fill cursor array. Writes ptr[n] = total.
__global__ __launch_bounds__(1024) void scan_kernel(uint32* __restrict__ cnt,
                                                    uint32* __restrict__ ptr, int n) {
  __shared__ uint32 wsum[32];
  __shared__ uint32 carry_s;
  const int tid = threadIdx.x;
  const int lane = tid & 31;
  const int wid = tid >> 5;
  if (tid == 0) carry_s = 0u;
  __syncthreads();
  for (int base = 0; base < n; base += 1024) {
    const int idx = base + tid;
    uint32 orig = (idx < n) ? cnt[idx] : 0u;
    uint32 x = orig;
    for (int off = 1; off < 32; off <<= 1) {       // inclusive wave scan
      uint32 y = __shfl_up(x, off, 32);
      if (lane >= off) x += y;
    }
    if (lane == 31) wsum[wid] = x;
    __syncthreads();
    if (wid == 0) {                                 // scan the 32 wave totals
      uint32 w = wsum[lane];
      for (int off = 1; off < 32; off <<= 1) {
        uint32 y = __shfl_up(w, off, 32);
        if (lane >= off) w += y;
      }
      wsum[lane] = w;
    }
    __syncthreads();
    uint32 waveoff = (wid == 0) ? 0u : wsum[wid - 1];
    uint32 carry = carry_s;
    if (idx < n) {
      ptr[idx] = carry + waveoff + (x - orig);      // exclusive
      cnt[idx] = 0u;                                // reset for fill pass
    }
    __syncthreads();
    if (tid == 0) carry_s += wsum[31];
    __syncthreads();
  }
  if (threadIdx.x == 0) ptr[n] = carry_s;
}

// scatter edges into CSR slots
__global__ void fill_kernel(const long long* __restrict__ erow, const long long* __restrict__ ecol,
                            const float* __restrict__ evals, const uint32* __restrict__ row_ptr,
                            uint32* __restrict__ cursor, uint32* __restrict__ csr_col,
                            float* __restrict__ csr_val, int ne) {
  int i = blockIdx.x * blockDim.x + threadIdx.x;
  int stride = gridDim.x * blockDim.x;
  for (; i < ne; i += stride) {
    uint32 r = (uint32)erow[i];
    uint32 pos = row_ptr[r] + atomicAdd(&cursor[r], 1u);
    csr_col[pos] = (uint32)ecol[i];
    csr_val[pos] = evals[i];
  }
}

// ---------------- host side ----------------
extern "C" void kernel_launch(void* const* d_in, const int* in_sizes, int n_in,
                              void* d_out, int out_size, void* d_ws, size_t ws_size,
                              hipStream_t stream) {
  (void)n_in; (void)out_size; (void)ws_size;
  const float*     users = (const float*)d_in[0];
  const long long* erow  = (const long long*)d_in[1];
  const long long* ecol  = (const long long*)d_in[2];
  const float*     evals = (const float*)d_in[3];
  const int n_emb   = in_sizes[0];        // N * 128
  const int n_nodes = n_emb / 128;
  const int n_edges = in_sizes[1];
  float* out = (float*)d_out;

  // workspace layout (float4 users stay 16B-aligned)
  float*  emb_a   = (float*)d_ws;
  float*  emb_b   = emb_a + (size_t)n_emb;
  uint32* row_ptr = (uint32*)(emb_b + (size_t)n_emb);
  uint32* cursor  = row_ptr + (size_t)(n_nodes + 1);
  uint32* csr_col = cursor + (size_t)n_nodes;
  float*  csr_val = (float*)(csr_col + (size_t)n_edges);

  const int n4 = n_emb / 4;
  int gb_e  = (n_edges + BLOCKT - 1) / BLOCKT; if (gb_e  > 8192) gb_e  = 8192;
  int gb_n4 = (n4      + BLOCKT - 1) / BLOCKT; if (gb_n4 > 8192) gb_n4 = 8192;
  int gb_c  = (n_nodes + BLOCKT - 1) / BLOCKT;

  // build CSR once
  zero_u32_kernel<<<gb_c, BLOCKT, 0, stream>>>(cursor, n_nodes);
  init_out_kernel<<<gb_n4, BLOCKT, 0, stream>>>((const float4*)users, (float4*)out, n4);
  hist_kernel<<<gb_e, BLOCKT, 0, stream>>>(erow, cursor, n_edges);
  scan_kernel<<<1, 1024, 0, stream>>>(cursor, row_ptr, n_nodes);
  fill_kernel<<<gb_e, BLOCKT, 0, stream>>>(erow, ecol, evals, row_ptr, cursor,
                                           csr_col, csr_val, n_edges);

  // 3 hops, fused accumulate (+0.25x) into out; last hop skips dst store
  const int gb_s = 2048;
  spmm_csr_kernel<<<gb_s, BLOCKT, 0, stream>>>(row_ptr, csr_col, csr_val, users, emb_a, out, n_nodes);
  spmm_csr_kernel<<<gb_s, BLOCKT, 0, stream>>>(row_ptr, csr_col, csr_val, emb_a, emb_b, out, n_nodes);
  spmm_csr_kernel<<<gb_s, BLOCKT, 0, stream>>>(row_ptr, csr_col, csr_val, emb_b, nullptr, out, n_nodes);
}